// BlackwellLinear_1417339208448
// MI455X (gfx1250) — compile-verified
//
#include <hip/hip_runtime.h>

// ---------------------------------------------------------------------------
// CDNA5 (gfx1250, wave32) 2:4-sparse + int8-fake-quant linear:
//   out = x @ quant(sparse24(W)).T + bias
// V_SWMMAC_F32_16X16X64_F16: A = sparse weight (16 outs x K64, stored half),
// B = x tile (K64 x 16 tokens, f16) staged in LDS, C/D = 16x16 f32.
// Main path: x pre-converted to f16; B tile DMA'd by the Tensor Data Mover
// (one tensor_load_to_lds per 128-K chunk, HW-applied LDS padding, TENSORcnt).
// ---------------------------------------------------------------------------

typedef __attribute__((ext_vector_type(32))) _Float16 v32h;
typedef __attribute__((ext_vector_type(16))) _Float16 v16h;
typedef __attribute__((ext_vector_type(8)))  _Float16 v8h;
typedef __attribute__((ext_vector_type(8)))  float    v8f;
typedef __attribute__((ext_vector_type(4)))  unsigned u32x4;
typedef __attribute__((ext_vector_type(8)))  int      i32x8;
typedef __attribute__((ext_vector_type(4)))  int      i32x4;

#define STR64   72               // 64 data halves + 8 pad (KC=64 kernels)
#define TILE64  (64 * STR64)
#define STR128  136              // 128 data halves + 8 pad (TDM kernel)
#define TILE128 (64 * STR128)

// ---------------------------------------------------------------------------
// Pass 1: global max|W| (group max always survives the 2:4 mask).
// ---------------------------------------------------------------------------
__global__ __launch_bounds__(256) void k_absmax(const float* __restrict__ w,
                                                unsigned* __restrict__ scale_bits,
                                                int ngroups) {
  int g = blockIdx.x * blockDim.x + threadIdx.x;
  float m = 0.0f;
  if (g < ngroups) {
    const float4 v = reinterpret_cast<const float4*>(w)[g];
    m = fmaxf(fmaxf(fabsf(v.x), fabsf(v.y)), fmaxf(fabsf(v.z), fabsf(v.w)));
  }
  #pragma unroll
  for (int off = 16; off > 0; off >>= 1)
    m = fmaxf(m, __shfl_xor(m, off, 32));
  if ((threadIdx.x & 31) == 0)
    atomicMax(scale_bits, __float_as_uint(m));
}

// ---------------------------------------------------------------------------
// Pass 2: per (row, 32-K chunk): keep 2 largest-|.| of each group of 4,
// fake-quantize (RNE), emit 16 packed f16 + u32 metadata (ISA 7.12.4 layout).
// ---------------------------------------------------------------------------
__global__ __launch_bounds__(256) void k_pack24(const float* __restrict__ w,
                                                const unsigned* __restrict__ scale_bits,
                                                const int* __restrict__ precision,
                                                _Float16* __restrict__ pw,
                                                unsigned* __restrict__ meta,
                                                int nchunks, int in_f) {
  int c = blockIdx.x * blockDim.x + threadIdx.x;
  if (c >= nchunks) return;
  const int chunksPerRow = in_f >> 5;
  const int row   = c / chunksPerRow;
  const int chunk = c % chunksPerRow;

  const float qmax = (float)((1 << (precision[0] - 1)) - 1);
  float scale = __uint_as_float(scale_bits[0]) / qmax;
  if (!(scale > 0.0f)) scale = 1.0f;
  const float inv = 1.0f / scale;

  const float* wp = w + (size_t)row * in_f + chunk * 32;
  unsigned mm = 0;
  v16h pk;
  #pragma unroll
  for (int g = 0; g < 8; ++g) {
    const float4 v = *reinterpret_cast<const float4*>(wp + g * 4);
    const float f[4] = {v.x, v.y, v.z, v.w};
    const float a[4] = {fabsf(v.x), fabsf(v.y), fabsf(v.z), fabsf(v.w)};
    int b0 = 0;
    #pragma unroll
    for (int i = 1; i < 4; ++i) if (a[i] > a[b0]) b0 = i;
    int b1 = (b0 == 0) ? 1 : 0;
    #pragma unroll
    for (int i = 0; i < 4; ++i) if (i != b0 && a[i] > a[b1]) b1 = i;
    const int lo = min(b0, b1), hi = max(b0, b1);
    const float q0 = fminf(fmaxf(rintf(f[lo] * inv), -qmax), qmax) * scale;
    const float q1 = fminf(fmaxf(rintf(f[hi] * inv), -qmax), qmax) * scale;
    pk[2 * g]     = (_Float16)q0;
    pk[2 * g + 1] = (_Float16)q1;
    mm |= (unsigned)(lo | (hi << 2)) << (4 * g);
  }
  *reinterpret_cast<v16h*>(pw + (size_t)row * (in_f >> 1) + chunk * 16) = pk;
  meta[(size_t)row * chunksPerRow + chunk] = mm;
}

// ---------------------------------------------------------------------------
// Pass 2b (big-ws path): x f32 -> f16, 8 elems/thread.
// ---------------------------------------------------------------------------
__global__ __launch_bounds__(256) void k_xcvt(const float* __restrict__ x,
                                              _Float16* __restrict__ xh, size_t n) {
  size_t i = ((size_t)blockIdx.x * 256 + threadIdx.x) * 8;
  if (i >= n) return;
  const float4 f0 = *reinterpret_cast<const float4*>(x + i);
  const float4 f1 = *reinterpret_cast<const float4*>(x + i + 4);
  v8h h;
  h[0] = (_Float16)f0.x; h[1] = (_Float16)f0.y; h[2] = (_Float16)f0.z; h[3] = (_Float16)f0.w;
  h[4] = (_Float16)f1.x; h[5] = (_Float16)f1.y; h[6] = (_Float16)f1.z; h[7] = (_Float16)f1.w;
  *reinterpret_cast<v8h*>(xh + i) = h;
}

// ---------------------------------------------------------------------------
// Async copy 16B global -> LDS (ASYNCcnt-tracked).
// ---------------------------------------------------------------------------
__device__ __forceinline__ void async_ld16(unsigned lds_off, const _Float16* g) {
  asm volatile("global_load_async_to_lds_b128 %0, %1, off"
               :: "v"(lds_off), "v"(g) : "memory");
}
__device__ __forceinline__ void wait_async0() {
  asm volatile("s_wait_asynccnt 0" ::: "memory");
}

// ---------------------------------------------------------------------------
// B fragment load + swmmac core (one-fragment look-ahead so ds loads of
// fragment t+1 are in flight while swmmac t executes).
// ---------------------------------------------------------------------------
template<int STR>
__device__ __forceinline__ v32h loadB(const _Float16* sbuf, int t, int soff, int col, int hi) {
  const _Float16* brow = sbuf + (t * 16 + col) * STR + soff + hi * 16;
  const v8h b0 = *reinterpret_cast<const v8h*>(brow);
  const v8h b1 = *reinterpret_cast<const v8h*>(brow + 8);
  const v8h b2 = *reinterpret_cast<const v8h*>(brow + 32);
  const v8h b3 = *reinterpret_cast<const v8h*>(brow + 40);
  v32h b;
  #pragma unroll
  for (int i = 0; i < 8; ++i) {
    b[i] = b0[i]; b[8 + i] = b1[i]; b[16 + i] = b2[i]; b[24 + i] = b3[i];
  }
  return b;
}

template<int STR>
__device__ __forceinline__ void swmmac_step(const _Float16* sbuf, int soff, int col, int hi,
                                            const v16h& a, unsigned idx, v8f acc[4]) {
  v32h bcur = loadB<STR>(sbuf, 0, soff, col, hi);
  #pragma unroll
  for (int t = 0; t < 4; ++t) {
    v32h bnext;
    if (t < 3) bnext = loadB<STR>(sbuf, t + 1, soff, col, hi);
    acc[t] = __builtin_amdgcn_swmmac_f32_16x16x64_f16(false, a, false, bcur,
                                                      acc[t], (int)idx, false, false);
    bcur = bnext;
  }
}

__device__ __forceinline__ v16h loadA(const _Float16* ap, int ks) {
  const v8h alo = *reinterpret_cast<const v8h*>(ap + ks * 32);
  const v8h ahi = *reinterpret_cast<const v8h*>(ap + ks * 32 + 16);
  v16h a;
  #pragma unroll
  for (int i = 0; i < 8; ++i) { a[i] = alo[i]; a[i + 8] = ahi[i]; }
  return a;
}

__device__ __forceinline__ void epilogue(v8f acc[4], const float* bias, float* out,
                                         int tokBase, int col, int hi, int oTile, int out_f) {
  const float4 b0 = *reinterpret_cast<const float4*>(bias + oTile + hi * 8);
  const float4 b1 = *reinterpret_cast<const float4*>(bias + oTile + hi * 8 + 4);
  #pragma unroll
  for (int t = 0; t < 4; ++t) {
    float* op = out + (size_t)(tokBase + t * 16 + col) * out_f + oTile + hi * 8;
    const float4 s0 = {acc[t][0] + b0.x, acc[t][1] + b0.y, acc[t][2] + b0.z, acc[t][3] + b0.w};
    const float4 s1 = {acc[t][4] + b1.x, acc[t][5] + b1.y, acc[t][6] + b1.z, acc[t][7] + b1.w};
    *reinterpret_cast<float4*>(op)     = s0;
    *reinterpret_cast<float4*>(op + 4) = s1;
  }
}

// ---------------------------------------------------------------------------
// GEMM, TDM path: one tensor_load_to_lds per 128-K chunk fetches the whole
// 64-row x 256B x-tile, with LDS padding (stride 272B) applied by the DMA:
//   pad_enable=1, pad_interval=5 (64 DWORDs), pad_amount=3 (4 DWORDs).
// D# per ISA 8.3/8.4; issued by wave 0, synced via TENSORcnt + barrier.
// This toolchain's builtin takes 6 args (g0:u32x4, g1:i32x8, i32x4, i32x4,
// i32x8, cpol) -- groups 2/3 unused for 2D tensors, passed as zeros.
// ---------------------------------------------------------------------------
__global__ __launch_bounds__(256) void k_gemm_tdm(const _Float16* __restrict__ xh,
                                                  const _Float16* __restrict__ pw,
                                                  const unsigned* __restrict__ meta,
                                                  const float* __restrict__ bias,
                                                  float* __restrict__ out,
                                                  int in_f, int out_f) {
  __shared__ _Float16 sb[2][TILE128];
  const int lane = threadIdx.x & 31;
  const int wave = threadIdx.x >> 5;
  const int col  = lane & 15;
  const int hi   = lane >> 4;
  const int oTile   = blockIdx.x * 128 + wave * 16;
  const int tokBase = blockIdx.y * 64;
  const int halfK      = in_f >> 1;
  const int metaStride = in_f >> 5;
  const int nkc        = in_f >> 7;  // 128-K chunks

  const unsigned sbase = (unsigned)(size_t)(&sb[0][0]);
  const unsigned long long gbase =
      (unsigned long long)(size_t)(xh + (size_t)tokBase * in_f);

  auto issue_tdm = [&](int kc, int buf) {
    const unsigned lds = sbase + (unsigned)(buf * TILE128 * 2);
    const unsigned long long ga = gbase + (unsigned long long)kc * 256;  // bytes
    u32x4 g0;
    g0[0] = 1u;                                   // count=1, user D#
    g0[1] = lds;                                  // lds_addr
    g0[2] = (unsigned)ga;                         // global_addr[31:0]
    g0[3] = (unsigned)(ga >> 32) | (2u << 30);    // global_addr[56:32] | type=2
    i32x8 g1;
    g1[0] = (1 << 16) | (1 << 20) | (5 << 22) | (3 << 25);  // 2B, pad 4dw/64dw
    g1[1] = (in_f & 0xFFFF) << 16;                // tensor_dim0[15:0]
    g1[2] = (in_f >> 16) | (64 << 16);            // dim0[31:16] | tensor_dim1=64
    g1[3] = 128 << 16;                            // dim1[31:16]=0 | tile_dim0=128
    g1[4] = 64;                                   // tile_dim1=64, tile_dim2=0
    g1[5] = in_f;                                 // tensor_dim0_stride[31:0]
    g1[6] = 0;                                    // stride[47:32] | dim1_stride lo
    g1[7] = 0;
    const i32x4 gz4 = {0, 0, 0, 0};
    const i32x8 gz8 = {0, 0, 0, 0, 0, 0, 0, 0};
    __builtin_amdgcn_tensor_load_to_lds(g0, g1, gz4, gz4, gz8, 0);
  };

  if (wave == 0) {
    issue_tdm(0, 0);
    __builtin_amdgcn_s_wait_tensorcnt(0);
  }
  __syncthreads();

  const _Float16* ap = pw + (size_t)(oTile + col) * halfK + hi * 8;
  const unsigned* mp = meta + (size_t)(oTile + col) * metaStride;
  v8f acc[4] = {v8f{}, v8f{}, v8f{}, v8f{}};

  for (int kc = 0; kc < nkc; ++kc) {
    const int cur = kc & 1;
    if (wave == 0 && kc + 1 < nkc) issue_tdm(kc + 1, cur ^ 1);

    #pragma unroll
    for (int s = 0; s < 2; ++s) {                 // two K64 sub-steps per chunk
      const int ks = kc * 2 + s;
      const v16h a = loadA(ap, ks);
      const unsigned idx = mp[ks * 2 + hi];
      swmmac_step<STR128>(&sb[cur][0], s * 64, col, hi, a, idx, acc);
    }
    if (wave == 0 && kc + 1 < nkc) __builtin_amdgcn_s_wait_tensorcnt(0);
    __syncthreads();
  }
  epilogue(acc, bias, out, tokBase, col, hi, oTile, out_f);
}

// ---------------------------------------------------------------------------
// GEMM, async-DMA path (kept as alternate): per-thread async b128 fills.
// ---------------------------------------------------------------------------
__global__ __launch_bounds__(256) void k_gemm_async(const _Float16* __restrict__ xh,
                                                    const _Float16* __restrict__ pw,
                                                    const unsigned* __restrict__ meta,
                                                    const float* __restrict__ bias,
                                                    float* __restrict__ out,
                                                    int in_f, int out_f) {
  __shared__ _Float16 sb[2][TILE64];
  const int lane = threadIdx.x & 31;
  const int wave = threadIdx.x >> 5;
  const int col  = lane & 15;
  const int hi   = lane >> 4;
  const int oTile   = blockIdx.x * 128 + wave * 16;
  const int tokBase = blockIdx.y * 64;
  const int halfK      = in_f >> 1;
  const int metaStride = in_f >> 5;
  const int nks        = in_f >> 6;

  const int frow = threadIdx.x >> 2;
  const int fc   = (threadIdx.x & 3) * 16;
  const _Float16* gx = xh + (size_t)(tokBase + frow) * in_f + fc;
  const unsigned sbase = (unsigned)(size_t)(&sb[0][0]);
  const unsigned ldst  = sbase + (unsigned)((frow * STR64 + fc) * 2);

  async_ld16(ldst, gx);
  async_ld16(ldst + 16, gx + 8);
  wait_async0();
  __syncthreads();

  const _Float16* ap = pw + (size_t)(oTile + col) * halfK + hi * 8;
  const unsigned* mp = meta + (size_t)(oTile + col) * metaStride;
  v8f acc[4] = {v8f{}, v8f{}, v8f{}, v8f{}};

  for (int ks = 0; ks < nks; ++ks) {
    const int cur = ks & 1;
    if (ks + 1 < nks) {
      const unsigned dst = sbase + (unsigned)((cur ^ 1) * TILE64 * 2) +
                           (unsigned)((frow * STR64 + fc) * 2);
      async_ld16(dst, gx + (ks + 1) * 64);
      async_ld16(dst + 16, gx + (ks + 1) * 64 + 8);
    }
    const v16h a = loadA(ap, ks);
    const unsigned idx = mp[ks * 2 + hi];
    swmmac_step<STR64>(&sb[cur][0], 0, col, hi, a, idx, acc);
    if (ks + 1 < nks) wait_async0();
    __syncthreads();
  }
  epilogue(acc, bias, out, tokBase, col, hi, oTile, out_f);
}

// ---------------------------------------------------------------------------
// GEMM, fallback path (small ws): cooperative f32->f16 conversion into LDS.
// ---------------------------------------------------------------------------
__global__ __launch_bounds__(256) void k_gemm_cvt(const float* __restrict__ x,
                                                  const _Float16* __restrict__ pw,
                                                  const unsigned* __restrict__ meta,
                                                  const float* __restrict__ bias,
                                                  float* __restrict__ out,
                                                  int in_f, int out_f) {
  __shared__ _Float16 sb[2][TILE64];
  const int lane = threadIdx.x & 31;
  const int wave = threadIdx.x >> 5;
  const int col  = lane & 15;
  const int hi   = lane >> 4;
  const int oTile   = blockIdx.x * 128 + wave * 16;
  const int tokBase = blockIdx.y * 64;
  const int halfK      = in_f >> 1;
  const int metaStride = in_f >> 5;
  const int nks        = in_f >> 6;

  const int frow = threadIdx.x >> 2;
  const int fc   = (threadIdx.x & 3) * 16;
  const float* gx = x + (size_t)(tokBase + frow) * in_f + fc;
  _Float16* sdst0 = &sb[0][frow * STR64 + fc];
  _Float16* sdst1 = &sb[1][frow * STR64 + fc];

  {
    const float4 f0 = *reinterpret_cast<const float4*>(gx);
    const float4 f1 = *reinterpret_cast<const float4*>(gx + 4);
    const float4 f2 = *reinterpret_cast<const float4*>(gx + 8);
    const float4 f3 = *reinterpret_cast<const float4*>(gx + 12);
    v8h h0, h1;
    h0[0]=(_Float16)f0.x; h0[1]=(_Float16)f0.y; h0[2]=(_Float16)f0.z; h0[3]=(_Float16)f0.w;
    h0[4]=(_Float16)f1.x; h0[5]=(_Float16)f1.y; h0[6]=(_Float16)f1.z; h0[7]=(_Float16)f1.w;
    h1[0]=(_Float16)f2.x; h1[1]=(_Float16)f2.y; h1[2]=(_Float16)f2.z; h1[3]=(_Float16)f2.w;
    h1[4]=(_Float16)f3.x; h1[5]=(_Float16)f3.y; h1[6]=(_Float16)f3.z; h1[7]=(_Float16)f3.w;
    *reinterpret_cast<v8h*>(sdst0)     = h0;
    *reinterpret_cast<v8h*>(sdst0 + 8) = h1;
  }
  __syncthreads();

  const _Float16* ap = pw + (size_t)(oTile + col) * halfK + hi * 8;
  const unsigned* mp = meta + (size_t)(oTile + col) * metaStride;
  v8f acc[4] = {v8f{}, v8f{}, v8f{}, v8f{}};

  for (int ks = 0; ks < nks; ++ks) {
    const int cur = ks & 1;
    float4 f0, f1, f2, f3;
    if (ks + 1 < nks) {
      const float* g = gx + (ks + 1) * 64;
      f0 = *reinterpret_cast<const float4*>(g);
      f1 = *reinterpret_cast<const float4*>(g + 4);
      f2 = *reinterpret_cast<const float4*>(g + 8);
      f3 = *reinterpret_cast<const float4*>(g + 12);
    }
    const v16h a = loadA(ap, ks);
    const unsigned idx = mp[ks * 2 + hi];
    swmmac_step<STR64>(&sb[cur][0], 0, col, hi, a, idx, acc);
    if (ks + 1 < nks) {
      v8h h0, h1;
      h0[0]=(_Float16)f0.x; h0[1]=(_Float16)f0.y; h0[2]=(_Float16)f0.z; h0[3]=(_Float16)f0.w;
      h0[4]=(_Float16)f1.x; h0[5]=(_Float16)f1.y; h0[6]=(_Float16)f1.z; h0[7]=(_Float16)f1.w;
      h1[0]=(_Float16)f2.x; h1[1]=(_Float16)f2.y; h1[2]=(_Float16)f2.z; h1[3]=(_Float16)f2.w;
      h1[4]=(_Float16)f3.x; h1[5]=(_Float16)f3.y; h1[6]=(_Float16)f3.z; h1[7]=(_Float16)f3.w;
      _Float16* d = (cur ? sdst0 : sdst1);
      *reinterpret_cast<v8h*>(d)     = h0;
      *reinterpret_cast<v8h*>(d + 8) = h1;
    }
    __syncthreads();
  }
  epilogue(acc, bias, out, tokBase, col, hi, oTile, out_f);
}

// ---------------------------------------------------------------------------
// Launch. ws: [0..4) scale | [256..) packed W f16 | meta u32 | (opt) x f16.
// ---------------------------------------------------------------------------
extern "C" void kernel_launch(void* const* d_in, const int* in_sizes, int n_in,
                              void* d_out, int out_size, void* d_ws, size_t ws_size,
                              hipStream_t stream) {
  const float* x    = (const float*)d_in[0];
  const float* w    = (const float*)d_in[1];
  const float* bias = (const float*)d_in[2];
  const int*   prec = (const int*)d_in[3];
  float*       out  = (float*)d_out;

  const int out_f  = in_sizes[2];
  const int in_f   = in_sizes[1] / out_f;
  const int tokens = in_sizes[0] / in_f;

  unsigned char* ws = (unsigned char*)d_ws;
  unsigned* scale_bits = (unsigned*)ws;
  _Float16* pw = (_Float16*)(ws + 256);
  const size_t pwBytes   = (size_t)out_f * (in_f / 2) * sizeof(_Float16);
  unsigned* meta = (unsigned*)(ws + 256 + pwBytes);
  const size_t metaBytes = (size_t)out_f * (in_f / 32) * sizeof(unsigned);
  const size_t xhOff   = (256 + pwBytes + metaBytes + 4095) & ~(size_t)4095;
  const size_t xhBytes = (size_t)tokens * in_f * sizeof(_Float16);
  _Float16* xh = (_Float16*)(ws + xhOff);

  const int ngroups4  = (out_f * in_f) / 4;
  const int nchunks32 = (out_f * in_f) / 32;

  (void)hipMemsetAsync(scale_bits, 0, sizeof(unsigned), stream);
  k_absmax<<<(ngroups4 + 255) / 256, 256, 0, stream>>>(w, scale_bits, ngroups4);
  k_pack24<<<(nchunks32 + 255) / 256, 256, 0, stream>>>(w, scale_bits, prec, pw, meta,
                                                        nchunks32, in_f);

  dim3 grid(out_f / 128, tokens / 64);
  if (ws_size >= xhOff + xhBytes && (in_f % 128) == 0) {
    const size_t nx = (size_t)tokens * in_f;
    k_xcvt<<<(int)((nx / 8 + 255) / 256), 256, 0, stream>>>(x, xh, nx);
    k_gemm_tdm<<<grid, 256, 0, stream>>>(xh, pw, meta, bias, out, in_f, out_f);
  } else {
    k_gemm_cvt<<<grid, 256, 0, stream>>>(x, pw, meta, bias, out, in_f, out_f);
  }
}